// LocalSphereAttention_25125558681855
// MI455X (gfx1250) — compile-verified
//
#include <hip/hip_runtime.h>
#include <hip/hip_bf16.h>
#include <math.h>

// ---------------- problem constants ----------------
#define BATCH   4
#define NPTS    4096
#define DIMC    256
#define NHEADS  8
#define HEADD   32
#define KNN     32

typedef __attribute__((ext_vector_type(2))) float v2f;
typedef __attribute__((ext_vector_type(8))) float v8f;

__device__ __forceinline__ v8f wmma_f32(v2f a, v2f b, v8f c) {
    return __builtin_amdgcn_wmma_f32_16x16x4_f32(
        /*neg_a=*/false, a, /*neg_b=*/false, b,
        /*c_mod=*/(short)0, c, /*reuse_a=*/false, /*reuse_b=*/false);
}

// =====================================================================
// 1) KNN: one thread per (b,n), 32-entry sorted insertion list.
//    Stable insertion (strict '>' while shifting) matches jax top_k tie
//    ordering (lower index first among equal distances).
// =====================================================================
__global__ void lsa_knn_kernel(const float* __restrict__ xyz,
                               int* __restrict__ knn_idx) {
    int gid = blockIdx.x * blockDim.x + threadIdx.x;      // b*NPTS + n
    int b = gid >> 12;                                     // /4096
    int n = gid & (NPTS - 1);
    const float* base = xyz + (size_t)b * NPTS * 3;
    float qx = base[n * 3 + 0];
    float qy = base[n * 3 + 1];
    float qz = base[n * 3 + 2];

    float bd[KNN];
    int   bi[KNN];
#pragma unroll
    for (int j = 0; j < KNN; ++j) { bd[j] = INFINITY; bi[j] = -1; }

    for (int m = 0; m < NPTS; ++m) {
        float dx = qx - base[m * 3 + 0];
        float dy = qy - base[m * 3 + 1];
        float dz = qz - base[m * 3 + 2];
        float d2 = dx * dx + dy * dy + dz * dz;
        if (d2 < bd[KNN - 1]) {
            int p = KNN - 1;
            while (p > 0 && bd[p - 1] > d2) {
                bd[p] = bd[p - 1]; bi[p] = bi[p - 1]; --p;
            }
            bd[p] = d2; bi[p] = m;
        }
    }
#pragma unroll
    for (int j = 0; j < KNN; ++j) knn_idx[(size_t)gid * KNN + j] = bi[j];
}

// =====================================================================
// 2a) Fused Q/K/V projection, f32 WMMA (V_WMMA_F32_16X16X4_F32).
//     One wave owns a 32x16 output region (two 16x16 M-subtiles) of all
//     THREE outputs: 6 accumulators. Per K-step (K+=4):
//       2 x b64 A loads + 6 x b32 B loads -> 6 WMMAs
//     (A amortized over 3 weight matrices, B amortized over 2 M-subtiles).
//     Fragment layouts per CDNA5 ISA 7.12.2:
//       A (16x4):  lane -> M = lane&15 ; elem e -> K = 2*(lane>>4)+e
//       B (4x16):  lane -> N = lane&15 ; elem e -> K = 2*(lane>>4)+e
//       C (16x16): VGPR r -> M = r + 8*(lane>>4) ; N = lane&15
// =====================================================================
#define GEMM_WAVES_PER_BLOCK 8

__global__ void lsa_qkv_gemm_wmma(const float* __restrict__ X,
                                  const float* __restrict__ Wq, const float* __restrict__ bq,
                                  const float* __restrict__ Wk, const float* __restrict__ bk,
                                  const float* __restrict__ Wv, const float* __restrict__ bv,
                                  float* __restrict__ Q,
                                  float* __restrict__ Kf,
                                  float* __restrict__ Vf,
                                  int M) {
    const int NN = DIMC;
    int wave = threadIdx.x >> 5;
    int lane = threadIdx.x & 31;
    int tile = blockIdx.x * GEMM_WAVES_PER_BLOCK + wave;   // 32-row x 16-col tiles
    const int ntilesN = NN / 16;                           // 16
    int m0 = (tile / ntilesN) * 32;
    int n0 = (tile % ntilesN) * 16;
    if (m0 >= M) return;                                   // wave-uniform exit

    int ml = lane & 15;
    int kh = lane >> 4;

    const float* xrow0 = X + (size_t)(m0 + ml) * DIMC;
    const float* xrow1 = xrow0 + (size_t)16 * DIMC;
    const float* wq = Wq + n0 + ml;
    const float* wk = Wk + n0 + ml;
    const float* wv = Wv + n0 + ml;

    __builtin_prefetch(xrow0, 0, 3);   // global_prefetch_b8
    __builtin_prefetch(xrow1, 0, 3);

    v8f cq0 = {}, cq1 = {}, ck0 = {}, ck1 = {}, cv0 = {}, cv1 = {};
#pragma unroll 2
    for (int k0 = 0; k0 < DIMC; k0 += 4) {
        int kb = k0 + 2 * kh;
        v2f a0, a1, fq, fk, fv;
        a0.x = xrow0[kb];  a0.y = xrow0[kb + 1];
        a1.x = xrow1[kb];  a1.y = xrow1[kb + 1];
        fq.x = wq[(size_t)kb * NN];  fq.y = wq[(size_t)(kb + 1) * NN];
        fk.x = wk[(size_t)kb * NN];  fk.y = wk[(size_t)(kb + 1) * NN];
        fv.x = wv[(size_t)kb * NN];  fv.y = wv[(size_t)(kb + 1) * NN];
        cq0 = wmma_f32(a0, fq, cq0);
        cq1 = wmma_f32(a1, fq, cq1);
        ck0 = wmma_f32(a0, fk, ck0);
        ck1 = wmma_f32(a1, fk, ck1);
        cv0 = wmma_f32(a0, fv, cv0);
        cv1 = wmma_f32(a1, fv, cv1);
    }

    float vbq = bq[n0 + ml];
    float vbk = bk[n0 + ml];
    float vbv = bv[n0 + ml];
    size_t col = (size_t)n0 + ml;
#pragma unroll
    for (int r = 0; r < 8; ++r) {
        size_t row0 = (size_t)m0 + 8 * kh + r;
        size_t row1 = row0 + 16;
        Q [row0 * NN + col] = cq0[r] + vbq;
        Q [row1 * NN + col] = cq1[r] + vbq;
        Kf[row0 * NN + col] = ck0[r] + vbk;
        Kf[row1 * NN + col] = ck1[r] + vbk;
        Vf[row0 * NN + col] = cv0[r] + vbv;
        Vf[row1 * NN + col] = cv1[r] + vbv;
    }
}

// =====================================================================
// 2b) Single-weight GEMM (output projection), same 2x M-subtile scheme:
//     per K-step: 2 x b64 A + 2 x b32 B -> 2 WMMAs.
// =====================================================================
__global__ void lsa_gemm256_wmma(const float* __restrict__ X,
                                 const float* __restrict__ W,
                                 const float* __restrict__ bias,
                                 float* __restrict__ Y,
                                 int M) {
    const int NN = DIMC;
    int wave = threadIdx.x >> 5;
    int lane = threadIdx.x & 31;
    int tile = blockIdx.x * GEMM_WAVES_PER_BLOCK + wave;   // 32-row x 16-col tiles
    const int ntilesN = NN / 16;
    int m0 = (tile / ntilesN) * 32;
    int n0 = (tile % ntilesN) * 16;
    if (m0 >= M) return;

    int ml = lane & 15;
    int kh = lane >> 4;

    const float* xrow0 = X + (size_t)(m0 + ml) * DIMC;
    const float* xrow1 = xrow0 + (size_t)16 * DIMC;
    const float* wcol  = W + n0 + ml;

    __builtin_prefetch(xrow0, 0, 3);
    __builtin_prefetch(xrow1, 0, 3);

    v8f c0 = {}, c1 = {};
#pragma unroll 2
    for (int k0 = 0; k0 < DIMC; k0 += 4) {
        int kb = k0 + 2 * kh;
        v2f a0, a1, bm;
        a0.x = xrow0[kb];  a0.y = xrow0[kb + 1];
        a1.x = xrow1[kb];  a1.y = xrow1[kb + 1];
        bm.x = wcol[(size_t)kb * NN];
        bm.y = wcol[(size_t)(kb + 1) * NN];
        c0 = wmma_f32(a0, bm, c0);
        c1 = wmma_f32(a1, bm, c1);
    }

    float bv = bias[n0 + ml];
    size_t col = (size_t)n0 + ml;
#pragma unroll
    for (int r = 0; r < 8; ++r) {
        size_t row0 = (size_t)m0 + 8 * kh + r;
        Y[row0 * NN + col]        = c0[r] + bv;
        Y[(row0 + 16) * NN + col] = c1[r] + bv;
    }
}

// =====================================================================
// 3) Local attention: one 256-thread block per (b,n).
//    Stage A: load idx + q; gather 32 neighbor K rows into LDS
//             (pitch 257 floats -> conflict-free: bank = (j + c) % 64);
//             lanes 0..31 also run the 3->32(relu)->8 bias MLP.
//    Stage B: wave w == head h computes 32 scores (lane = neighbor),
//             softmax via __shfl_xor tree reductions (wave32).
//    Stage C: reuse the same LDS buffer for the V gather.
//    Stage D: thread t = h*32+dh accumulates sum_j attn[h][j]*V[j][t].
// =====================================================================
#define NB_PITCH 257   // 257 mod 64 == 1 -> neighbor index maps to distinct banks

__global__ void lsa_attn_kernel(const float* __restrict__ Q,
                                const float* __restrict__ Kf,
                                const float* __restrict__ Vf,
                                const float* __restrict__ xyz,
                                const int*   __restrict__ knn_idx,
                                const float* __restrict__ W1,
                                const float* __restrict__ b1,
                                const float* __restrict__ W2,
                                const float* __restrict__ b2,
                                float* __restrict__ P) {
    __shared__ float snb[KNN][NB_PITCH];   // K rows, then reused for V rows
    __shared__ float sq[DIMC];
    __shared__ float sb[KNN][NHEADS];      // position bias per (neighbor, head)
    __shared__ float sa[NHEADS][KNN];      // attention weights
    __shared__ int   sidx[KNN];

    int bn = blockIdx.x;                   // b*NPTS + n
    int b  = bn >> 12;
    int n  = bn & (NPTS - 1);
    int t  = threadIdx.x;

    if (t < KNN) sidx[t] = knn_idx[(size_t)bn * KNN + t];
    sq[t] = Q[(size_t)bn * DIMC + t];
    __syncthreads();

    // gather neighbor K rows (coalesced across t)
    const size_t bbase = (size_t)b * NPTS;
#pragma unroll 4
    for (int j = 0; j < KNN; ++j)
        snb[j][t] = Kf[(bbase + sidx[j]) * DIMC + t];

    // bias MLP on lanes 0..31 (thread j -> neighbor j)
    if (t < KNN) {
        const float* xb = xyz + bbase * 3;
        float rx = xb[n * 3 + 0] - xb[sidx[t] * 3 + 0];
        float ry = xb[n * 3 + 1] - xb[sidx[t] * 3 + 1];
        float rz = xb[n * 3 + 2] - xb[sidx[t] * 3 + 2];
        float hid[32];
#pragma unroll
        for (int i = 0; i < 32; ++i) {
            float v = rx * W1[i] + ry * W1[32 + i] + rz * W1[64 + i] + b1[i];
            hid[i] = v > 0.0f ? v : 0.0f;
        }
#pragma unroll
        for (int h = 0; h < NHEADS; ++h) {
            float acc = b2[h];
#pragma unroll
            for (int i = 0; i < 32; ++i) acc += hid[i] * W2[i * NHEADS + h];
            sb[t][h] = acc;
        }
    }
    __syncthreads();

    // scores + softmax: wave w = head h, lane = neighbor j
    {
        int h = t >> 5;
        int j = t & 31;
        float s = 0.0f;
#pragma unroll
        for (int d = 0; d < HEADD; ++d)
            s += sq[h * HEADD + d] * snb[j][h * HEADD + d];
        s = s * 0.17677669529663687f + sb[j][h];   // 1/sqrt(32)

        float mx = s;
        for (int off = 16; off > 0; off >>= 1)
            mx = fmaxf(mx, __shfl_xor(mx, off, 32));
        float e = __expf(s - mx);
        float sum = e;
        for (int off = 16; off > 0; off >>= 1)
            sum += __shfl_xor(sum, off, 32);
        sa[h][j] = e / sum;
    }
    __syncthreads();

    // reuse snb for V rows
#pragma unroll 4
    for (int j = 0; j < KNN; ++j)
        snb[j][t] = Vf[(bbase + sidx[j]) * DIMC + t];
    __syncthreads();

    // weighted V sum: t == h*32+dh
    {
        int h = t >> 5;
        float o = 0.0f;
#pragma unroll
        for (int j = 0; j < KNN; ++j)
            o += sa[h][j] * snb[j][t];
        P[(size_t)bn * DIMC + t] = o;
    }
}

// =====================================================================
// launch
// =====================================================================
extern "C" void kernel_launch(void* const* d_in, const int* in_sizes, int n_in,
                              void* d_out, int out_size, void* d_ws, size_t ws_size,
                              hipStream_t stream) {
    (void)in_sizes; (void)n_in; (void)out_size; (void)ws_size;

    const float* x   = (const float*)d_in[0];
    const float* xyz = (const float*)d_in[1];
    const float* Wq  = (const float*)d_in[2];
    const float* bq  = (const float*)d_in[3];
    const float* Wk  = (const float*)d_in[4];
    const float* bk  = (const float*)d_in[5];
    const float* Wv  = (const float*)d_in[6];
    const float* bv  = (const float*)d_in[7];
    const float* Wo  = (const float*)d_in[8];
    const float* bo  = (const float*)d_in[9];
    const float* W1  = (const float*)d_in[10];
    const float* b1  = (const float*)d_in[11];
    const float* W2  = (const float*)d_in[12];
    const float* b2  = (const float*)d_in[13];
    float* out = (float*)d_out;

    const size_t MN = (size_t)BATCH * NPTS;            // 16384 rows
    float* ws = (float*)d_ws;
    float* Q  = ws;
    float* Kf = Q  + MN * DIMC;
    float* Vf = Kf + MN * DIMC;
    float* P  = Vf + MN * DIMC;
    int* knn_idx = (int*)(P + MN * DIMC);

    // 1) KNN indices
    lsa_knn_kernel<<<(int)(MN / 256), 256, 0, stream>>>(xyz, knn_idx);

    // 2) fused Q/K/V projection (f32 WMMA, 32x16 tiles per wave, 3 outputs)
    const int gemmBlocks = (int)((MN / 32) * (DIMC / 16) / GEMM_WAVES_PER_BLOCK);
    lsa_qkv_gemm_wmma<<<gemmBlocks, 32 * GEMM_WAVES_PER_BLOCK, 0, stream>>>(
        x, Wq, bq, Wk, bk, Wv, bv, Q, Kf, Vf, (int)MN);

    // 3) local attention with positional bias MLP
    lsa_attn_kernel<<<(int)MN, 256, 0, stream>>>(Q, Kf, Vf, xyz, knn_idx,
                                                 W1, b1, W2, b2, P);

    // 4) output projection (f32 WMMA) -> d_out
    lsa_gemm256_wmma<<<gemmBlocks, 32 * GEMM_WAVES_PER_BLOCK, 0, stream>>>(
        P, Wo, bo, out, (int)MN);
}